// Net_54142357733422
// MI455X (gfx1250) — compile-verified
//
#include <hip/hip_runtime.h>
#include <math.h>

// CDNA5 / gfx1250: wave32, WMMA 16x16x4 F32 for the dense root GEMM epilogues.
typedef __attribute__((ext_vector_type(2))) float v2f;
typedef __attribute__((ext_vector_type(8))) float v8f;

// ---------------------------------------------------------------- fill
__global__ void fill_kernel(float* __restrict__ p, long long n, float v) {
    long long i = (long long)blockIdx.x * blockDim.x + threadIdx.x;
    long long stride = (long long)gridDim.x * blockDim.x;
    for (; i < n; i += stride) p[i] = v;
}

// ---------------------------------------------------------------- per-(dst, relation) edge counts
__global__ void count_kernel(const int* __restrict__ dst, const int* __restrict__ et,
                             float* __restrict__ cnt, int E, int R) {
    int e = blockIdx.x * blockDim.x + threadIdx.x;
    if (e >= E) return;
    atomicAdd(&cnt[dst[e] * R + et[e]], 1.0f);
}

// ---------------------------------------------------------------- mean-aggr edge pass, in=3 (block 1->2), out=6
// msg[b*2+o] = xin[src,b] * W[et][b*2+o], normalized by cnt[dst*R+et]
__global__ void edge_mean_1to2(const int* __restrict__ src, const int* __restrict__ dst,
                               const int* __restrict__ et, const float* __restrict__ xin,
                               const float* __restrict__ W, const float* __restrict__ cnt,
                               float* __restrict__ agg, int E, int R) {
    int e = blockIdx.x * blockDim.x + threadIdx.x;
    if (e >= E) return;
    int s = src[e], d = dst[e], t = et[e];
    float inv = 1.0f / fmaxf(cnt[d * R + t], 1.0f);
    const float* w = W + t * 6;                 // [NB=3][in_b=1][out_b=2]
    float x0 = xin[s * 3 + 0] * inv;
    float x1 = xin[s * 3 + 1] * inv;
    float x2 = xin[s * 3 + 2] * inv;
    float* a = agg + d * 6;
    atomicAdd(a + 0, x0 * w[0]); atomicAdd(a + 1, x0 * w[1]);
    atomicAdd(a + 2, x1 * w[2]); atomicAdd(a + 3, x1 * w[3]);
    atomicAdd(a + 4, x2 * w[4]); atomicAdd(a + 5, x2 * w[5]);
}

// ---------------------------------------------------------------- add-aggr edge pass, in=6 (block 2->1), out=3
// msg[b] = h[src,2b]*W[et][b*2+0] + h[src,2b+1]*W[et][b*2+1]
__global__ void edge_add_2to1(const int* __restrict__ src, const int* __restrict__ dst,
                              const int* __restrict__ et, const float* __restrict__ hin,
                              const float* __restrict__ W, float* __restrict__ agg, int E) {
    int e = blockIdx.x * blockDim.x + threadIdx.x;
    if (e >= E) return;
    int s = src[e], d = dst[e], t = et[e];
    const float* w = W + t * 6;                 // [NB=3][in_b=2][out_b=1]
    const float* h = hin + s * 6;
    float m0 = h[0] * w[0] + h[1] * w[1];
    float m1 = h[2] * w[2] + h[3] * w[3];
    float m2 = h[4] * w[4] + h[5] * w[5];
    float* a = agg + d * 3;
    atomicAdd(a + 0, m0); atomicAdd(a + 1, m1); atomicAdd(a + 2, m2);
}

// ---------------------------------------------------------------- WMMA node epilogue
// h[n,:] = relu(agg[n,:] + xin[n,:] @ root + bias), one wave per 16 nodes.
// A (16xK f32, K padded to mult of 4), B (Kx16 = root zero-padded),
// C = agg tile + bias.  V_WMMA_F32_16X16X4_F32 layouts (ISA 7.12.2):
//   A 16x4: lane<16 -> (M=lane, K=0/1), lane>=16 -> (M=lane-16, K=2/3)
//   B 4x16: lane<16 -> (K=0/1, N=lane), lane>=16 -> (K=2/3, N=lane-16)
//   C/D   : vgpr r, lane<16 -> (M=r, N=lane); lane>=16 -> (M=r+8, N=lane-16)
// All lane-varying conditions are clamped-address loads + selects -> EXEC
// stays all-ones at the WMMA (required by ISA).  Early-out is wave-uniform.
template <int IN, int OUT>
__global__ __launch_bounds__(256)
void node_wmma(const float* __restrict__ xin, const float* __restrict__ agg,
               const float* __restrict__ root, const float* __restrict__ bias,
               float* __restrict__ hout, int nNodes) {
    const int wave = (int)((blockIdx.x * blockDim.x + threadIdx.x) >> 5);
    const int lane = threadIdx.x & 31;
    const int base = wave * 16;
    if (base >= nNodes) return;                 // wave-uniform
    const int half = lane >> 4;                 // 0: M/K low half, 1: high half
    const int m = lane & 15;                    // A-row / B,C,D-column
    const int mc = (m < OUT) ? m : (OUT - 1);   // clamped column (garbage cols discarded)
    const float bv = bias[mc];

    v8f c;
#pragma unroll
    for (int r = 0; r < 8; ++r) {
        int row = base + half * 8 + r;
        c[r] = agg[row * OUT + mc] + bv;        // cols >= OUT hold finite garbage, never stored
    }

    // K-chunk 0 (K = 0..3)
    {
        int k0 = half * 2, k1 = half * 2 + 1;
        float a0 = xin[(base + m) * IN + ((k0 < IN) ? k0 : 0)];
        float a1 = xin[(base + m) * IN + ((k1 < IN) ? k1 : 0)];
        float b0 = root[((k0 < IN) ? k0 : 0) * OUT + mc];
        float b1 = root[((k1 < IN) ? k1 : 0) * OUT + mc];
        v2f a, b;
        a.x = (k0 < IN) ? a0 : 0.0f; a.y = (k1 < IN) ? a1 : 0.0f;
        b.x = (k0 < IN) ? b0 : 0.0f; b.y = (k1 < IN) ? b1 : 0.0f;
        c = __builtin_amdgcn_wmma_f32_16x16x4_f32(false, a, false, b, (short)0, c,
                                                  false, false);
    }
    // K-chunk 1 (K = 4..7) only when IN > 4 (layer 2, IN=6)
    if (IN > 4) {
        int k0 = 4 + half * 2, k1 = 5 + half * 2;
        float a0 = xin[(base + m) * IN + ((k0 < IN) ? k0 : 0)];
        float a1 = xin[(base + m) * IN + ((k1 < IN) ? k1 : 0)];
        float b0 = root[((k0 < IN) ? k0 : 0) * OUT + mc];
        float b1 = root[((k1 < IN) ? k1 : 0) * OUT + mc];
        v2f a, b;
        a.x = (k0 < IN) ? a0 : 0.0f; a.y = (k1 < IN) ? a1 : 0.0f;
        b.x = (k0 < IN) ? b0 : 0.0f; b.y = (k1 < IN) ? b1 : 0.0f;
        c = __builtin_amdgcn_wmma_f32_16x16x4_f32(false, a, false, b, (short)0, c,
                                                  false, false);
    }

#pragma unroll
    for (int r = 0; r < 8; ++r) {
        if (m < OUT) {
            int row = base + half * 8 + r;
            hout[row * OUT + m] = fmaxf(c[r], 0.0f);
        }
    }
}

// ---------------------------------------------------------------- global pool (sum of h[:,0..5])
__global__ void pool_kernel(const float* __restrict__ h, float* __restrict__ pooled, int n) {
    float acc[6] = {0.f, 0.f, 0.f, 0.f, 0.f, 0.f};
    for (int i = blockIdx.x * blockDim.x + threadIdx.x; i < n;
         i += gridDim.x * blockDim.x) {
#pragma unroll
        for (int j = 0; j < 6; ++j) acc[j] += h[i * 6 + j];
    }
#pragma unroll
    for (int j = 0; j < 6; ++j)
        for (int off = 16; off > 0; off >>= 1) acc[j] += __shfl_down(acc[j], off, 32);
    if ((threadIdx.x & 31) == 0) {
#pragma unroll
        for (int j = 0; j < 6; ++j) atomicAdd(&pooled[j], acc[j]);
    }
}

// ---------------------------------------------------------------- mean + log_softmax
__global__ void finish_kernel(const float* __restrict__ pooled, float* __restrict__ out,
                              float invN) {
    if (blockIdx.x == 0 && threadIdx.x == 0) {
        float v[6], mx = -1e30f;
        for (int j = 0; j < 6; ++j) { v[j] = pooled[j] * invN; mx = fmaxf(mx, v[j]); }
        float s = 0.0f;
        for (int j = 0; j < 6; ++j) s += expf(v[j] - mx);
        float l = logf(s);
        for (int j = 0; j < 6; ++j) out[j] = v[j] - mx - l;
    }
}

extern "C" void kernel_launch(void* const* d_in, const int* in_sizes, int n_in,
                              void* d_out, int out_size, void* d_ws, size_t ws_size,
                              hipStream_t stream) {
    const float* x     = (const float*)d_in[0];
    const int*   eidx  = (const int*)d_in[1];
    const int*   et    = (const int*)d_in[3];
    const float* W1    = (const float*)d_in[4];
    const float* root1 = (const float*)d_in[5];
    const float* b1    = (const float*)d_in[6];
    const float* W2    = (const float*)d_in[7];
    const float* root2 = (const float*)d_in[8];
    const float* b2    = (const float*)d_in[9];
    const float* W3    = (const float*)d_in[10];
    const float* root3 = (const float*)d_in[11];
    const float* b3    = (const float*)d_in[12];

    const int N = in_sizes[0] / 3;
    const int E = in_sizes[3];
    const int R = in_sizes[4] / 6;          // W1: [R,3,1,2]
    const int* src = eidx;                  // edge_index[0]
    const int* dst = eidx + E;              // edge_index[1]

    // Workspace layout (floats): cnt[N*R] | agg[N*6] | hA[N*6] | hB[N*3] | pooled[8]
    float* ws     = (float*)d_ws;
    float* cnt    = ws;
    float* agg    = cnt + (long long)N * R;
    float* hA     = agg + (long long)N * 6;   // h1, then reused for h3
    float* hB     = hA + (long long)N * 6;    // h2
    float* pooled = hB + (long long)N * 3;

    const int TB = 256;
    const int edgeBlocks = (E + TB - 1) / TB;
    const int waves = (N + 15) / 16;
    const int nodeBlocks = (waves * 32 + TB - 1) / TB;

    // Zero-init scratch (harness poisons once; we must re-zero per call).
    fill_kernel<<<2048, TB, 0, stream>>>(cnt, (long long)N * R, 0.0f);
    fill_kernel<<<256, TB, 0, stream>>>(agg, (long long)N * 6, 0.0f);
    fill_kernel<<<1, 32, 0, stream>>>(pooled, 8, 0.0f);

    // Edge counts per (dst, relation) — shared by mean layers 1 and 3.
    count_kernel<<<edgeBlocks, TB, 0, stream>>>(dst, et, cnt, E, R);

    // ---- Layer 1: 3 -> 6, mean aggr
    edge_mean_1to2<<<edgeBlocks, TB, 0, stream>>>(src, dst, et, x, W1, cnt, agg, E, R);
    node_wmma<3, 6><<<nodeBlocks, TB, 0, stream>>>(x, agg, root1, b1, hA, N);

    // ---- Layer 2: 6 -> 3, add aggr
    fill_kernel<<<256, TB, 0, stream>>>(agg, (long long)N * 3, 0.0f);
    edge_add_2to1<<<edgeBlocks, TB, 0, stream>>>(src, dst, et, hA, W2, agg, E);
    node_wmma<6, 3><<<nodeBlocks, TB, 0, stream>>>(hA, agg, root2, b2, hB, N);

    // ---- Layer 3: 3 -> 6, mean aggr
    fill_kernel<<<256, TB, 0, stream>>>(agg, (long long)N * 6, 0.0f);
    edge_mean_1to2<<<edgeBlocks, TB, 0, stream>>>(src, dst, et, hB, W3, cnt, agg, E, R);
    node_wmma<3, 6><<<nodeBlocks, TB, 0, stream>>>(hB, agg, root3, b3, hA, N);

    // ---- Global mean pool + log_softmax
    pool_kernel<<<400, TB, 0, stream>>>(hA, pooled, N);
    finish_kernel<<<1, 32, 0, stream>>>(pooled, (float*)d_out, 1.0f / (float)N);
}